// AffCoeffToMatrix_13958643712058
// MI455X (gfx1250) — compile-verified
//
#include <hip/hip_runtime.h>
#include <stdint.h>

#define TILE 256

// ---------------- CDNA5 async LDS<->global data movers (gfx1250) ----------------

__device__ __forceinline__ void async_copy_b128(float* lds_dst, const float* gsrc) {
#if defined(__gfx1250__)
  uint32_t l = (uint32_t)(uintptr_t)lds_dst;        // low 32 bits of flat addr == LDS byte offset
  uint64_t g = (uint64_t)(uintptr_t)gsrc;
  asm volatile("global_load_async_to_lds_b128 %0, %1, off th:TH_LOAD_NT"
               :: "v"(l), "v"(g) : "memory");
#else
  *(float4*)lds_dst = *(const float4*)gsrc;
#endif
}

__device__ __forceinline__ void async_copy_b32(float* lds_dst, const float* gsrc) {
#if defined(__gfx1250__)
  uint32_t l = (uint32_t)(uintptr_t)lds_dst;
  uint64_t g = (uint64_t)(uintptr_t)gsrc;
  asm volatile("global_load_async_to_lds_b32 %0, %1, off th:TH_LOAD_NT"
               :: "v"(l), "v"(g) : "memory");
#else
  *lds_dst = *gsrc;
#endif
}

__device__ __forceinline__ void async_store_b128(float* gdst, const float* lds_src) {
#if defined(__gfx1250__)
  uint32_t l = (uint32_t)(uintptr_t)lds_src;
  uint64_t g = (uint64_t)(uintptr_t)gdst;
  asm volatile("global_store_async_from_lds_b128 %0, %1, off th:TH_STORE_NT"
               :: "v"(g), "v"(l) : "memory");
#else
  *(float4*)gdst = *(const float4*)lds_src;
#endif
}

__device__ __forceinline__ void wait_async0() {
#if defined(__gfx1250__)
  asm volatile("s_wait_asynccnt 0" ::: "memory");
#endif
}

// ---------------- math ----------------

// Rodrigues: R = I + a*K + b*K^2, K = skew(v), K^2 = v v^T - |v|^2 I
// a = sin(t)/t, b = (1-cos t)/t^2, small-angle limits a=1, b=0.5 (matches reference).
__device__ __forceinline__ void rodrigues3(float x, float y, float z, float R[9]) {
  float th2 = __builtin_fmaf(x, x, __builtin_fmaf(y, y, z * z));
  float th  = __builtin_sqrtf(th2);
  float s   = __sinf(th);
  float c   = __cosf(th);
  float a   = s * __builtin_amdgcn_rcpf(th);
  float b   = (1.0f - c) * __builtin_amdgcn_rcpf(th2);
  bool small = th < 1e-6f;
  a = small ? 1.0f : a;
  b = small ? 0.5f : b;

  float bxy = b * x * y, bxz = b * x * z, byz = b * y * z;
  float ax = a * x, ay = a * y, az = a * z;
  R[0] = 1.0f + b * (x * x - th2);
  R[1] = bxy - az;
  R[2] = bxz + ay;
  R[3] = bxy + az;
  R[4] = 1.0f + b * (y * y - th2);
  R[5] = byz - ax;
  R[6] = bxz - ay;
  R[7] = byz + ax;
  R[8] = 1.0f + b * (z * z - th2);
}

// ---------------- kernel ----------------

__global__ __launch_bounds__(TILE) void affine_coeff_to_matrix(
    const float* __restrict__ trans, const float* __restrict__ rotat,
    const float* __restrict__ sdir,  const float* __restrict__ scal,
    float* __restrict__ out, long long Btot)
{
  // Separate input staging (4 x TILE*3 floats = 12 KB) and output staging
  // (TILE*12 floats = 12 KB) regions: no input-region reuse -> one less barrier.
  __shared__ __align__(16) float s_in[4 * 3 * TILE];
  __shared__ __align__(16) float s_out[12 * TILE];

  const int tid = threadIdx.x;
  const long long base = (long long)blockIdx.x * TILE;
  long long remll = Btot - base;
  const int n = (remll < (long long)TILE) ? (int)remll : TILE;
  if (n <= 0) return;

  const int nflt = n * 3;          // floats per input array this tile (<= 768)
  const int nf4  = nflt >> 2;      // whole float4 chunks (<= 192 < TILE)
  const int rem  = nflt & 3;       // ragged tail floats (< 4)

  const float* gin[4] = { trans + base * 3, rotat + base * 3,
                          sdir  + base * 3, scal  + base * 3 };

  // One guarded DMA issue per array per thread -- no loops needed since nf4 < TILE.
#pragma unroll
  for (int a = 0; a < 4; ++a) {
    float* l = &s_in[a * (3 * TILE)];
    const float* g = gin[a];
    if (tid < nf4) async_copy_b128(l + tid * 4, g + tid * 4);
    if (tid < rem) async_copy_b32(l + nf4 * 4 + tid, g + nf4 * 4 + tid);
  }
  wait_async0();        // this wave's async DMAs landed in LDS
  __syncthreads();      // all waves' DMAs visible

  if (tid < n) {
    const float tr0 = s_in[0 * 768 + tid * 3 + 0];
    const float tr1 = s_in[0 * 768 + tid * 3 + 1];
    const float tr2 = s_in[0 * 768 + tid * 3 + 2];
    const float ro0 = s_in[1 * 768 + tid * 3 + 0];
    const float ro1 = s_in[1 * 768 + tid * 3 + 1];
    const float ro2 = s_in[1 * 768 + tid * 3 + 2];
    const float sd0 = s_in[2 * 768 + tid * 3 + 0];
    const float sd1 = s_in[2 * 768 + tid * 3 + 1];
    const float sd2 = s_in[2 * 768 + tid * 3 + 2];
    const float sc0 = s_in[3 * 768 + tid * 3 + 0];
    const float sc1 = s_in[3 * 768 + tid * 3 + 1];
    const float sc2 = s_in[3 * 768 + tid * 3 + 2];

    float U[9];
    rodrigues3(sd0, sd1, sd2, U);
    const float d0 = __expf(sc0);
    const float d1 = __expf(sc1);
    const float d2 = __expf(sc2);

    // M1 = U * diag(d) * U^T (symmetric)
    float M1[9];
#pragma unroll
    for (int i = 0; i < 3; ++i) {
#pragma unroll
      for (int j = 0; j <= i; ++j) {
        float v = U[i * 3 + 0] * (d0 * U[j * 3 + 0])
                + U[i * 3 + 1] * (d1 * U[j * 3 + 1])
                + U[i * 3 + 2] * (d2 * U[j * 3 + 2]);
        M1[i * 3 + j] = v;
        M1[j * 3 + i] = v;
      }
    }

    float R[9];
    rodrigues3(ro0, ro1, ro2, R);

    float res[12];
#pragma unroll
    for (int i = 0; i < 3; ++i) {
#pragma unroll
      for (int j = 0; j < 3; ++j) {
        res[i * 4 + j] = R[i * 3 + 0] * M1[0 * 3 + j]
                       + R[i * 3 + 1] * M1[1 * 3 + j]
                       + R[i * 3 + 2] * M1[2 * 3 + j];
      }
    }
    res[0 * 4 + 3] = tr0;
    res[1 * 4 + 3] = tr1;
    res[2 * 4 + 3] = tr2;

#pragma unroll
    for (int k = 0; k < 12; ++k)
      s_out[tid * 12 + k] = res[k];
  }
  __syncthreads();      // output tile fully staged in LDS

  // Stream the tile out: n*12 floats is always a multiple of 4 and
  // n*3 <= 768 = 3*TILE float4 chunks -> exactly 3 guarded DMA issues.
  float* gout = out + base * 12;
  const int of4 = n * 3;
#pragma unroll
  for (int k = 0; k < 3; ++k) {
    const int i = tid + k * TILE;
    if (i < of4) async_store_b128(gout + i * 4, &s_out[i * 4]);
  }
  wait_async0();
}

// ---------------- launcher ----------------

extern "C" void kernel_launch(void* const* d_in, const int* in_sizes, int n_in,
                              void* d_out, int out_size, void* d_ws, size_t ws_size,
                              hipStream_t stream) {
  const float* trans = (const float*)d_in[0];
  const float* rotat = (const float*)d_in[1];
  const float* sdir  = (const float*)d_in[2];
  const float* scal  = (const float*)d_in[3];
  float* out = (float*)d_out;

  const long long Btot = (long long)in_sizes[0] / 3;
  const int blocks = (int)((Btot + TILE - 1) / TILE);

  hipLaunchKernelGGL(affine_coeff_to_matrix, dim3(blocks), dim3(TILE), 0, stream,
                     trans, rotat, sdir, scal, out, Btot);
}